// GraphUnet_90701119356961
// MI455X (gfx1250) — compile-verified
//
#include <hip/hip_runtime.h>
#include <hip/hip_bf16.h>

typedef float v2f __attribute__((ext_vector_type(2)));
typedef float v8f __attribute__((ext_vector_type(8)));

#define EPSV 1e-5f
#define SIGV 10.0f

// ---------------- elementwise / reduction helpers ----------------

__global__ __launch_bounds__(256) void pool_k(const float* __restrict__ in,
                                              float* __restrict__ out,
                                              int C, int Nout) {
  int idx = blockIdx.x * 256 + threadIdx.x;
  if (idx >= C * Nout) return;
  int c = idx / Nout, j = idx - c * Nout;
  int Nin = Nout * 2;
  const float* r = in + (size_t)c * Nin;
  int p = 2 * j;
  float s = r[p];
  if (p - 1 >= 0) s += r[p - 1];
  if (p + 1 < Nin) s += r[p + 1];
  out[idx] = s * (1.0f / 3.0f);
}

__global__ __launch_bounds__(256) void repeat2_k(const float* __restrict__ in,
                                                 float* __restrict__ out,
                                                 int C, int Nin) {
  int idx = blockIdx.x * 256 + threadIdx.x;
  if (idx >= C * Nin) return;
  int c = idx / Nin, n = idx - c * Nin;
  float v = in[idx];
  float* orow = out + (size_t)c * Nin * 2;
  orow[2 * n] = v;
  orow[2 * n + 1] = v;
}

// per-row (3 rows) std with ddof=1 -> rinv = 1/(std+0.01)
__global__ __launch_bounds__(256) void rowstats_k(const float* __restrict__ X,
                                                  int N, float* __restrict__ rinv) {
  __shared__ float s1[256], s2[256];
  int c = blockIdx.x, t = threadIdx.x;
  const float* r = X + (size_t)c * N;
  float a = 0.f, b = 0.f;
  for (int n = t; n < N; n += 256) { float v = r[n]; a += v; b += v * v; }
  s1[t] = a; s2[t] = b; __syncthreads();
  for (int o = 128; o > 0; o >>= 1) {
    if (t < o) { s1[t] += s1[t + o]; s2[t] += s2[t + o]; }
    __syncthreads();
  }
  if (t == 0) {
    float mean = s1[0] / (float)N;
    float var = (s2[0] - (float)N * mean * mean) / (float)(N - 1);
    var = fmaxf(var, 0.f);
    rinv[c] = 1.0f / (sqrtf(var) + 0.01f);
  }
}

__global__ __launch_bounds__(256) void xn_sq_k(const float* __restrict__ X,
                                               const float* __restrict__ rinv,
                                               float* __restrict__ Xn,
                                               float* __restrict__ sq, int N) {
  int n = blockIdx.x * 256 + threadIdx.x;
  if (n >= N) return;
  float s = 0.f;
#pragma unroll
  for (int c = 0; c < 3; c++) {
    float v = X[c * N + n] * rinv[c];
    Xn[c * N + n] = v;
    s += v * v;
  }
  sq[n] = s;
}

__global__ __launch_bounds__(256) void we_k(const float* __restrict__ Xn,
                                            const float* __restrict__ sq,
                                            float* __restrict__ W, int N) {
  size_t idx = (size_t)blockIdx.x * 256 + threadIdx.x;
  int i = (int)(idx / (size_t)N);
  int j = (int)(idx - (size_t)i * N);
  float dot = Xn[i] * Xn[j] + Xn[N + i] * Xn[N + j] + Xn[2 * N + i] * Xn[2 * N + j];
  float D = fmaxf(sq[i] + sq[j] - 2.0f * dot, 0.0f);
  W[idx] = expf(-D * (1.0f / SIGV));
}

__global__ __launch_bounds__(256) void rowsum_k(const float* __restrict__ W,
                                                float* __restrict__ d, int N) {
  __shared__ float s[256];
  int i = blockIdx.x, t = threadIdx.x;
  const float* r = W + (size_t)i * N;
  float a = 0.f;
  for (int j = t; j < N; j += 256) a += r[j];
  s[t] = a; __syncthreads();
  for (int o = 128; o > 0; o >>= 1) {
    if (t < o) s[t] += s[t + o];
    __syncthreads();
  }
  if (t == 0) d[i] = s[0];
}

// L[i,j] = M[i]M[j]*((i==j ? d[i] : 0) - We[i,j]) + (i==j)*(1-M[i])  (already symmetric)
__global__ __launch_bounds__(256) void lapfin_k(float* __restrict__ W,
                                                const float* __restrict__ d,
                                                const float* __restrict__ m, int N) {
  size_t idx = (size_t)blockIdx.x * 256 + threadIdx.x;
  int i = (int)(idx / (size_t)N);
  int j = (int)(idx - (size_t)i * N);
  float w = W[idx];
  float v = m[i] * m[j] * (((i == j) ? d[i] : 0.0f) - w);
  if (i == j) v += 1.0f - m[i];
  W[idx] = v;
}

// ---------------- WMMA GEMM: Y[C x N] = A[C x N] @ B[N x N] ----------------
// Wave tile: 32(M) x 32(N): 2 A frags x 2 B frags -> 4 accumulators.
// A frag (16x4 f32): lane m = lane%16; vgpr0=K(2*hi), vgpr1=K(2*hi+1), hi=lane/16.
// B frag (4x16 f32): lane n = lane%16; vgpr0=row(2*hi), vgpr1=row(2*hi+1).
// C/D frag: vgpr r -> row r + 8*hi, col = lane%16.
// Requires: C % 32 == 0, N % 256 == 0 (block = 8 waves x 32 cols).

__global__ __launch_bounds__(256) void gemm_wmma_k(const float* __restrict__ A,
                                                   const float* __restrict__ B,
                                                   float* __restrict__ Y,
                                                   int C, int N) {
  int lane = threadIdx.x & 31;
  int wave = threadIdx.x >> 5;
  int n0 = (blockIdx.x * 8 + wave) * 32;
  int m0 = blockIdx.y * 32;
  int ml = lane & 15, hi = lane >> 4;

  const float* ar0 = A + (size_t)(m0 + ml) * N;
  const float* ar1 = ar0 + (size_t)16 * N;
  const float* bp = B + (size_t)(2 * hi) * N + n0 + ml;  // += 4N per k-step

  v8f acc00 = {0.f, 0.f, 0.f, 0.f, 0.f, 0.f, 0.f, 0.f};
  v8f acc01 = acc00, acc10 = acc00, acc11 = acc00;

#pragma unroll 2
  for (int k0 = 0; k0 < N; k0 += 4) {
    int ka = k0 + 2 * hi;
    v2f a0, a1, b0, b1;
    a0.x = ar0[ka];      a0.y = ar0[ka + 1];
    a1.x = ar1[ka];      a1.y = ar1[ka + 1];
    b0.x = bp[0];        b0.y = bp[N];
    b1.x = bp[16];       b1.y = bp[N + 16];
    acc00 = __builtin_amdgcn_wmma_f32_16x16x4_f32(false, a0, false, b0,
                                                  (short)0, acc00, false, false);
    acc01 = __builtin_amdgcn_wmma_f32_16x16x4_f32(false, a0, false, b1,
                                                  (short)0, acc01, false, false);
    acc10 = __builtin_amdgcn_wmma_f32_16x16x4_f32(false, a1, false, b0,
                                                  (short)0, acc10, false, false);
    acc11 = __builtin_amdgcn_wmma_f32_16x16x4_f32(false, a1, false, b1,
                                                  (short)0, acc11, false, false);
    bp += (size_t)4 * N;
  }

  float* y0 = Y + (size_t)(m0 + hi * 8) * N + n0 + ml;
  float* y1 = Y + (size_t)(m0 + 16 + hi * 8) * N + n0 + ml;
#pragma unroll
  for (int r = 0; r < 8; r++) {
    y0[(size_t)r * N] = acc00[r];
    y0[(size_t)r * N + 16] = acc01[r];
    y1[(size_t)r * N] = acc10[r];
    y1[(size_t)r * N + 16] = acc11[r];
  }
}

// ---------------- WMMA conv (width-9, pad 4) as 9 shifted GEMMs ----------------
// Z[o, n] = sum_t sum_k Wt[o*sM + k*sK + t] * Yin[k, n + tsign*(t-4)]
// forward:    sM = Cin*9,  sK = 9,        tsign = +1   (K[o,i,t])
// transposed: sM = 9,      sK = Cout*9,   tsign = -1
// Wave tile: 16(o) x 32(n): shared weight frag, 2 column frags -> 2 accumulators.
// Requires: Cout % 16 == 0, Cin % 4 == 0, N % 256 == 0.

__global__ __launch_bounds__(256) void conv_wmma_k(const float* __restrict__ Yin,
                                                   const float* __restrict__ Wt,
                                                   float* __restrict__ Z,
                                                   int Cout, int Cin, int N,
                                                   int sM, int sK, int tsign) {
  int lane = threadIdx.x & 31;
  int wave = threadIdx.x >> 5;
  int n0 = (blockIdx.x * 8 + wave) * 32;
  int o0 = blockIdx.y * 16;
  int ml = lane & 15, hi = lane >> 4;

  v8f acc0 = {0.f, 0.f, 0.f, 0.f, 0.f, 0.f, 0.f, 0.f};
  v8f acc1 = acc0;

  for (int t = 0; t < 9; t++) {
    int off = tsign * (t - 4);
    int p0 = n0 + ml + off;
    int p1 = p0 + 16;
    bool inb0 = (p0 >= 0) && (p0 < N);
    bool inb1 = (p1 >= 0) && (p1 < N);
    int pc0 = p0 < 0 ? 0 : (p0 >= N ? N - 1 : p0);
    int pc1 = p1 < 0 ? 0 : (p1 >= N ? N - 1 : p1);
    const float* bc0 = Yin + pc0;
    const float* bc1 = Yin + pc1;
    const float* wbase = Wt + (size_t)(o0 + ml) * sM + t;

    for (int k0 = 0; k0 < Cin; k0 += 4) {
      int kk = k0 + 2 * hi;
      v2f a, b0, b1;
      a.x = wbase[(size_t)kk * sK];
      a.y = wbase[(size_t)(kk + 1) * sK];
      float v00 = bc0[(size_t)kk * N];
      float v01 = bc0[(size_t)(kk + 1) * N];
      float v10 = bc1[(size_t)kk * N];
      float v11 = bc1[(size_t)(kk + 1) * N];
      b0.x = inb0 ? v00 : 0.0f;
      b0.y = inb0 ? v01 : 0.0f;
      b1.x = inb1 ? v10 : 0.0f;
      b1.y = inb1 ? v11 : 0.0f;
      acc0 = __builtin_amdgcn_wmma_f32_16x16x4_f32(false, a, false, b0,
                                                   (short)0, acc0, false, false);
      acc1 = __builtin_amdgcn_wmma_f32_16x16x4_f32(false, a, false, b1,
                                                   (short)0, acc1, false, false);
    }
  }

  float* zrow = Z + (size_t)(o0 + hi * 8) * N + n0 + ml;
#pragma unroll
  for (int r = 0; r < 8; r++) {
    zrow[(size_t)r * N] = acc0[r];
    zrow[(size_t)r * N + 16] = acc1[r];
  }
}

// ---------------- fused mask * instance_norm -> relu (+ residual/skip) ----------------
// out[c,n] = (base ? base[c,n] : 0) + relu((z - mu_c) * rsqrt(var_c + eps)),
// z = m[n] * Zc[c,n]; mu/var over row n (ddof=0). One block per channel. N % 256 == 0, N<=4096.

__global__ __launch_bounds__(256) void norm_k(const float* __restrict__ Zc,
                                              const float* __restrict__ mr,
                                              const float* __restrict__ base,
                                              float* __restrict__ out, int N) {
  __shared__ float s1[256], s2[256];
  int c = blockIdx.x, t = threadIdx.x;
  int nper = N >> 8;
  const float* zr = Zc + (size_t)c * N;
  float v[16];
  float a = 0.f, b = 0.f;
  for (int r = 0; r < nper; r++) {
    int n = r * 256 + t;
    float z = mr[n] * zr[n];
    v[r] = z; a += z; b += z * z;
  }
  s1[t] = a; s2[t] = b; __syncthreads();
  for (int o = 128; o > 0; o >>= 1) {
    if (t < o) { s1[t] += s1[t + o]; s2[t] += s2[t + o]; }
    __syncthreads();
  }
  float mu = s1[0] / (float)N;
  float var = s2[0] / (float)N - mu * mu;
  var = fmaxf(var, 0.f);
  float rs = rsqrtf(var + EPSV);
  const float* br = base ? base + (size_t)c * N : nullptr;
  float* orow = out + (size_t)c * N;
  for (int r = 0; r < nper; r++) {
    int n = r * 256 + t;
    float q = fmaxf((v[r] - mu) * rs, 0.f);
    orow[n] = (br ? br[n] : 0.f) + q;
  }
}

// ---------------- host orchestration ----------------

extern "C" void kernel_launch(void* const* d_in, const int* in_sizes, int n_in,
                              void* d_out, int out_size, void* d_ws, size_t ws_size,
                              hipStream_t stream) {
  (void)in_sizes; (void)n_in; (void)out_size; (void)ws_size;
  const float* x_in = (const float*)d_in[0];
  const float* X_in = (const float*)d_in[1];
  const float* m_in = (const float*)d_in[2];
  const float* Kp[11];
  for (int i = 0; i < 11; i++) Kp[i] = (const float*)d_in[3 + i];

  const int Ns[4] = {4096, 2048, 1024, 512};
  const int Cs[4] = {32, 64, 128, 256};

  float* wsf = (float*)d_ws;
  size_t off = 0;
  auto carve = [&](size_t nf) -> float* {
    float* p = wsf + off;
    off += (nf + 63) & ~(size_t)63;
    return p;
  };
  float* Lp[4];
  Lp[0] = carve(4096ull * 4096);
  Lp[1] = carve(2048ull * 2048);
  Lp[2] = carve(1024ull * 1024);
  Lp[3] = carve(512ull * 512);
  float* mlv[4]; for (int l = 0; l < 4; l++) mlv[l] = carve(Ns[l]);
  float* Xlv[4]; for (int l = 0; l < 4; l++) Xlv[l] = carve(3 * Ns[l]);
  float* Xn = carve(3 * 4096);
  float* sq = carve(4096);
  float* drow = carve(4096);
  float* rinv = carve(64);
  float* xbuf = carve(262144);
  float* ybuf = carve(262144);
  float* zbuf = carve(262144);
  float* xs[3]; for (int i = 0; i < 3; i++) xs[i] = carve(131072);

  hipMemcpyAsync(mlv[0], m_in, 4096 * sizeof(float), hipMemcpyDeviceToDevice, stream);
  hipMemcpyAsync(Xlv[0], X_in, 3 * 4096 * sizeof(float), hipMemcpyDeviceToDevice, stream);
  hipMemcpyAsync(xbuf, x_in, 32 * 4096 * sizeof(float), hipMemcpyDeviceToDevice, stream);

  // --- Precompute all 4 graph Laplacians (depend only on X, m) ---
  for (int l = 0; l < 4; l++) {
    int N = Ns[l];
    if (l > 0) {
      pool_k<<<dim3((N + 255) / 256), 256, 0, stream>>>(mlv[l - 1], mlv[l], 1, N);
      pool_k<<<dim3((3 * N + 255) / 256), 256, 0, stream>>>(Xlv[l - 1], Xlv[l], 3, N);
    }
    rowstats_k<<<3, 256, 0, stream>>>(Xlv[l], N, rinv);
    xn_sq_k<<<dim3((N + 255) / 256), 256, 0, stream>>>(Xlv[l], rinv, Xn, sq, N);
    unsigned nb = (unsigned)(((size_t)N * N) / 256);
    we_k<<<dim3(nb), 256, 0, stream>>>(Xn, sq, Lp[l], N);
    rowsum_k<<<N, 256, 0, stream>>>(Lp[l], drow, N);
    lapfin_k<<<dim3(nb), 256, 0, stream>>>(Lp[l], drow, mlv[l], N);
  }

  auto gemm = [&](const float* A, const float* B, float* Y, int C, int N) {
    gemm_wmma_k<<<dim3(N / 256, C / 32), 256, 0, stream>>>(A, B, Y, C, N);
  };
  auto conv = [&](const float* Yin, const float* W, float* Z, int Cout, int Cin,
                  int N, bool tr) {
    int sMv = tr ? 9 : Cin * 9;
    int sKv = tr ? Cout * 9 : 9;
    conv_wmma_k<<<dim3(N / 256, Cout / 16), 256, 0, stream>>>(
        Yin, W, Z, Cout, Cin, N, sMv, sKv, tr ? -1 : 1);
  };
  auto nrm = [&](const float* Z, const float* mr, const float* bs, float* out,
                 int C, int N) {
    norm_k<<<C, 256, 0, stream>>>(Z, mr, bs, out, N);
  };

  // --- Down path ---
  for (int l = 0; l < 3; l++) {
    int C = Cs[l], N = Ns[l];
    for (int s = 0; s < 2; s++) {  // smooth: x = x + relu(IN(m*conv(xL,K)))
      gemm(xbuf, Lp[l], ybuf, C, N);
      conv(ybuf, Kp[3 * l + s], zbuf, C, C, N, false);
      nrm(zbuf, mlv[l], xbuf, xbuf, C, N);
    }
    // coarsen: save skip, x = relu(IN(m*conv(xL,K))), pool
    hipMemcpyAsync(xs[l], xbuf, (size_t)C * N * sizeof(float),
                   hipMemcpyDeviceToDevice, stream);
    gemm(xbuf, Lp[l], ybuf, C, N);
    conv(ybuf, Kp[3 * l + 2], zbuf, 2 * C, C, N, false);
    nrm(zbuf, mlv[l], nullptr, zbuf, 2 * C, N);
    pool_k<<<dim3((2 * C * (N / 2) + 255) / 256), 256, 0, stream>>>(zbuf, xbuf,
                                                                    2 * C, N / 2);
  }
  // bottom smooths (forward conv): K9, K10 at C=256, N=512
  for (int s = 9; s <= 10; s++) {
    gemm(xbuf, Lp[3], ybuf, 256, 512);
    conv(ybuf, Kp[s], zbuf, 256, 256, 512, false);
    nrm(zbuf, mlv[3], xbuf, xbuf, 256, 512);
  }

  // --- Up path (transposed convs) ---
  for (int s = 10; s >= 9; s--) {
    gemm(xbuf, Lp[3], ybuf, 256, 512);
    conv(ybuf, Kp[s], zbuf, 256, 256, 512, true);
    nrm(zbuf, mlv[3], xbuf, xbuf, 256, 512);
  }
  for (int l = 2; l >= 0; l--) {
    int C = Cs[l], N = Ns[l];
    // refine with K[3l+2]: repeat, matmul with finer L, convT, relu + skip
    repeat2_k<<<dim3((2 * C * (N / 2) + 255) / 256), 256, 0, stream>>>(xbuf, ybuf,
                                                                       2 * C, N / 2);
    gemm(ybuf, Lp[l], zbuf, 2 * C, N);
    conv(zbuf, Kp[3 * l + 2], ybuf, C, 2 * C, N, true);
    nrm(ybuf, mlv[l], xs[l], xbuf, C, N);
    // smooths K[3l+1], K[3l]
    for (int s = 1; s >= 0; s--) {
      bool last = (l == 0 && s == 0);
      gemm(xbuf, Lp[l], ybuf, C, N);
      conv(ybuf, Kp[3 * l + s], zbuf, C, C, N, true);
      nrm(zbuf, mlv[l], xbuf, last ? (float*)d_out : xbuf, C, N);
    }
  }
}